// DeepGCN_Sta_12841952215495
// MI455X (gfx1250) — compile-verified
//
#include <hip/hip_runtime.h>
#include <math.h>

#define BB 4
#define NNP 4096
#define KK 16
#define BN (BB*NNP)            // 16384 points
#define CIN 64
#define INCH 67                // 3 xyz + 64 features
#define EDGES (BN*KK)          // 262144
#define BLOCKM 128
#define KC 64
#define KCP 66                 // padded LDS row stride (bank-conflict avoidance)
#define FUSED_C 384

typedef float v2f __attribute__((ext_vector_type(2)));
typedef float v8f __attribute__((ext_vector_type(8)));

// ---------------- assemble in67 (BN x 67) and pos (BN x 3) ----------------
__global__ void build_kernel(const float* __restrict__ feats,
                             const float* __restrict__ coords,
                             float* __restrict__ in67,
                             float* __restrict__ pos) {
  int i = blockIdx.x * blockDim.x + threadIdx.x;
  const int total = BN * INCH;
  for (; i < total; i += gridDim.x * blockDim.x) {
    int p = i / INCH, c = i % INCH;
    float v = (c < 3) ? coords[p * 4 + 1 + c] : feats[p * CIN + (c - 3)];
    in67[i] = v;
    if (c < 3) pos[p * 3 + c] = v;
  }
}

// ---------------- brute-force cosine KNN (top-16) + exp(-dist) ----------------
__global__ __launch_bounds__(256) void knn_kernel(const float* __restrict__ pos,
                                                  int* __restrict__ nng,
                                                  float* __restrict__ expw) {
  const int b = blockIdx.x >> 4;                       // 16 blocks per batch
  const int qi = ((blockIdx.x & 15) << 8) + threadIdx.x;
  const int qrow = b * NNP + qi;
  const float px = pos[qrow * 3 + 0], py = pos[qrow * 3 + 1], pz = pos[qrow * 3 + 2];
  const float qinv = 1.f / (sqrtf(px * px + py * py + pz * pz) + 1e-12f);
  const float qx = px * qinv, qy = py * qinv, qz = pz * qinv;
  const float qsq = qx * qx + qy * qy + qz * qz;

  float val[KK];
  int   nid[KK];
#pragma unroll
  for (int s = 0; s < KK; ++s) { val[s] = -3.0e38f; nid[s] = 0; }

  __shared__ float4 cs[256];
  for (int tile = 0; tile < NNP; tile += 256) {
    {
      int crow = b * NNP + tile + threadIdx.x;
      float cx = pos[crow * 3 + 0], cy = pos[crow * 3 + 1], cz = pos[crow * 3 + 2];
      float ci = 1.f / (sqrtf(cx * cx + cy * cy + cz * cz) + 1e-12f);
      cx *= ci; cy *= ci; cz *= ci;
      cs[threadIdx.x] = make_float4(cx, cy, cz, cx * cx + cy * cy + cz * cz);
    }
    __syncthreads();
    for (int j = 0; j < 256; ++j) {
      float4 c4 = cs[j];
      float d = 2.f * (qx * c4.x + qy * c4.y + qz * c4.z) - qsq - c4.w;
      if (d > val[KK - 1]) {
        float cv = d; int ci2 = tile + j;
#pragma unroll
        for (int s = 0; s < KK; ++s) {
          if (cv > val[s]) {
            float tv = val[s]; int ti = nid[s];
            val[s] = cv; nid[s] = ci2; cv = tv; ci2 = ti;
          }
        }
      }
    }
    __syncthreads();
  }
#pragma unroll
  for (int s = 0; s < KK; ++s) {
    int jr = b * NNP + nid[s];
    nng[qrow * KK + s] = jr;                           // global row index
    float dx = pos[jr * 3 + 0] - px, dy = pos[jr * 3 + 1] - py, dz = pos[jr * 3 + 2] - pz;
    float dist = sqrtf(dx * dx + dy * dy + dz * dz + 1e-12f);
    expw[qrow * KK + s] = expf(-dist);
  }
}

// ---------------- fused gather + FP32 WMMA GEMM ----------------
// MODE 0: per-channel sum / sumsq partials (BN statistics pass)
// MODE 1: recompute + BN + ReLU + *expw + max over K + residual -> fused slice
// MODE 2: recompute + BN + ReLU elementwise -> output (final conv)
template <int COUT, bool GATHER, int MODE>
__global__ __launch_bounds__(256)
void gemm_kernel(const float* __restrict__ Ain, int a_stride, int Kdim, int Cin,
                 const float* __restrict__ W,
                 const int* __restrict__ nnidx,
                 const float* __restrict__ expw,
                 const float* __restrict__ scsh,
                 const float* __restrict__ resid,
                 float* __restrict__ outp, int out_stride) {
  constexpr int NT = COUT / 16;
  constexpr int CP = COUT + 2;
  __shared__ float A_s[BLOCKM * KCP];
  __shared__ float B_s[KC * CP];
  __shared__ int rows_j[BLOCKM];
  __shared__ float slot[2 * 8 * COUT];

  const int tid = threadIdx.x;
  const int blockM = blockIdx.x * BLOCKM;

  if (GATHER) {
    if (tid < BLOCKM) rows_j[tid] = nnidx[blockM + tid];
  }

  v8f acc[NT];
#pragma unroll
  for (int nt = 0; nt < NT; ++nt)
#pragma unroll
    for (int r = 0; r < 8; ++r) acc[nt][r] = 0.f;

  const int wv = tid >> 5;
  const int lane = tid & 31;
  const int half = lane >> 4;
  const int l16 = lane & 15;

  for (int k0 = 0; k0 < Kdim; k0 += KC) {
    __syncthreads();
    // stage A: msg[row][c] = c<Cin ? x_i[c] : x_j[c-Cin]-x_i[c-Cin]  (zero-pad)
    for (int l = tid; l < BLOCKM * KC; l += 256) {
      int r = l >> 6, cc = l & 63;
      int c = k0 + cc;
      float v = 0.f;
      if (c < Kdim) {
        if (GATHER) {
          int bn = (blockM + r) >> 4;
          if (c < Cin) v = Ain[bn * a_stride + c];
          else {
            int c2 = c - Cin;
            v = Ain[rows_j[r] * a_stride + c2] - Ain[bn * a_stride + c2];
          }
        } else {
          v = Ain[(blockM + r) * a_stride + c];
        }
      }
      A_s[r * KCP + cc] = v;
    }
    // stage B: transpose W (COUT x Kdim) -> B_s[k][cout]
    for (int l = tid; l < KC * COUT; l += 256) {
      int co = l >> 6, cc = l & 63;
      int c = k0 + cc;
      B_s[cc * CP + co] = (c < Kdim) ? W[co * Kdim + c] : 0.f;
    }
    __syncthreads();

    const float* Arow = &A_s[(wv * 16 + l16) * KCP];
    for (int kk = 0; kk < KC; kk += 4) {
      v2f a;                                            // ISA 16x4 f32 A layout
      a.x = Arow[kk + 2 * half];
      a.y = Arow[kk + 2 * half + 1];
#pragma unroll
      for (int nt = 0; nt < NT; ++nt) {
        v2f bf;                                         // mirrored 4x16 B layout
        bf.x = B_s[(kk + 2 * half) * CP + nt * 16 + l16];
        bf.y = B_s[(kk + 2 * half + 1) * CP + nt * 16 + l16];
        acc[nt] = __builtin_amdgcn_wmma_f32_16x16x4_f32(
            false, a, false, bf, (short)0, acc[nt], false, false);
      }
    }
  }

  if (MODE == 0) {
#pragma unroll
    for (int nt = 0; nt < NT; ++nt) {
      float s = 0.f, q = 0.f;
#pragma unroll
      for (int r = 0; r < 8; ++r) { float v = acc[nt][r]; s += v; q += v * v; }
      s += __shfl_xor(s, 16);
      q += __shfl_xor(q, 16);
      if (half == 0) {
        slot[wv * COUT + nt * 16 + l16] = s;
        slot[8 * COUT + wv * COUT + nt * 16 + l16] = q;
      }
    }
    __syncthreads();
    if (tid < COUT) {
      float s = 0.f, q = 0.f;
#pragma unroll
      for (int w2 = 0; w2 < 8; ++w2) {
        s += slot[w2 * COUT + tid];
        q += slot[8 * COUT + w2 * COUT + tid];
      }
      outp[blockIdx.x * COUT + tid] = s;
      outp[(gridDim.x + blockIdx.x) * COUT + tid] = q;
    }
  } else if (MODE == 1) {
    const int bn = (blockM >> 4) + wv;                  // this wave's point
#pragma unroll
    for (int nt = 0; nt < NT; ++nt) {
      int c = nt * 16 + l16;
      float sc = scsh[c], sh = scsh[COUT + c];
      float cmax = -3.0e38f;
#pragma unroll
      for (int r = 0; r < 8; ++r) {
        float v = fmaf(acc[nt][r], sc, sh);             // BN
        v = fmaxf(v, 0.f) * expw[bn * KK + r + 8 * half];
        cmax = fmaxf(cmax, v);
      }
      cmax = fmaxf(cmax, __shfl_xor(cmax, 16));         // max over all 16 neighbors
      if (half == 0) {
        float o = cmax;
        if (resid) o += resid[bn * FUSED_C + c];
        outp[bn * FUSED_C + c] = o;
      }
    }
  } else {
#pragma unroll
    for (int nt = 0; nt < NT; ++nt) {
      int c = nt * 16 + l16;
      float sc = scsh[c], sh = scsh[COUT + c];
#pragma unroll
      for (int r = 0; r < 8; ++r) {
        int m = blockM + wv * 16 + r + 8 * half;
        outp[m * out_stride + c] = fmaxf(fmaf(acc[nt][r], sc, sh), 0.f);
      }
    }
  }
}

// ---------------- reduce per-block partials -> scale/shift per channel ----------------
__global__ void stats_reduce(const float* __restrict__ part, int nblk, float invM,
                             const float* __restrict__ g, const float* __restrict__ be,
                             float* __restrict__ scsh, int cout) {
  __shared__ float sh_s[256], sh_q[256];
  int c = blockIdx.x;
  float s = 0.f, q = 0.f;
  for (int p = threadIdx.x; p < nblk; p += 256) {
    s += part[p * cout + c];
    q += part[(nblk + p) * cout + c];
  }
  sh_s[threadIdx.x] = s; sh_q[threadIdx.x] = q;
  __syncthreads();
  for (int off = 128; off > 0; off >>= 1) {
    if (threadIdx.x < off) {
      sh_s[threadIdx.x] += sh_s[threadIdx.x + off];
      sh_q[threadIdx.x] += sh_q[threadIdx.x + off];
    }
    __syncthreads();
  }
  if (threadIdx.x == 0) {
    float m = sh_s[0] * invM;
    float v = sh_q[0] * invM - m * m;
    float sc = g[c] * rsqrtf(v + 1e-5f);
    scsh[c] = sc;
    scsh[cout + c] = be[c] - m * sc;                    // bias cancels in BN
  }
}

extern "C" void kernel_launch(void* const* d_in, const int* in_sizes, int n_in,
                              void* d_out, int out_size, void* d_ws, size_t ws_size,
                              hipStream_t stream) {
  const float* feats  = (const float*)d_in[0];
  const float* coords = (const float*)d_in[1];
  // backbone_params flattened: (w,b,g,be) x 4, then out_params (w,b,g,be), batch_size
  const float* W1 = (const float*)d_in[2];  const float* g1 = (const float*)d_in[4];  const float* be1 = (const float*)d_in[5];
  const float* W2 = (const float*)d_in[6];  const float* g2 = (const float*)d_in[8];  const float* be2 = (const float*)d_in[9];
  const float* W3 = (const float*)d_in[10]; const float* g3 = (const float*)d_in[12]; const float* be3 = (const float*)d_in[13];
  const float* W4 = (const float*)d_in[14]; const float* g4 = (const float*)d_in[16]; const float* be4 = (const float*)d_in[17];
  const float* Wo = (const float*)d_in[18]; const float* go = (const float*)d_in[20]; const float* beo = (const float*)d_in[21];

  float* ws = (float*)d_ws;
  size_t o = 0;
  float* in67  = ws + o; o += (size_t)BN * INCH;
  float* pos   = ws + o; o += (size_t)BN * 3;
  float* fused = ws + o; o += (size_t)BN * FUSED_C;
  int*   nng   = (int*)(ws + o); o += (size_t)BN * KK;
  float* expw  = ws + o; o += (size_t)BN * KK;
  float* part  = ws + o; o += (size_t)2 * 2048 * 128;
  float* scsh  = ws + o; o += 256;
  float* out   = (float*)d_out;

  build_kernel<<<256, 256, 0, stream>>>(feats, coords, in67, pos);
  knn_kernel<<<BN / 256, 256, 0, stream>>>(pos, nng, expw);

  const int GE = EDGES / BLOCKM;               // 2048 blocks for edge GEMMs
  const float invME = 1.f / (float)EDGES;

  // Layer 1: 134 -> 64, no residual
  gemm_kernel<64, true, 0><<<GE, 256, 0, stream>>>(in67, INCH, 2 * INCH, INCH, W1, nng, expw, scsh, nullptr, part, 0);
  stats_reduce<<<64, 256, 0, stream>>>(part, GE, invME, g1, be1, scsh, 64);
  gemm_kernel<64, true, 1><<<GE, 256, 0, stream>>>(in67, INCH, 2 * INCH, INCH, W1, nng, expw, scsh, nullptr, fused + 0, FUSED_C);

  // Layer 2: 128 -> 64, residual (+feat1)
  gemm_kernel<64, true, 0><<<GE, 256, 0, stream>>>(fused + 0, FUSED_C, 128, 64, W2, nng, expw, scsh, nullptr, part, 0);
  stats_reduce<<<64, 256, 0, stream>>>(part, GE, invME, g2, be2, scsh, 64);
  gemm_kernel<64, true, 1><<<GE, 256, 0, stream>>>(fused + 0, FUSED_C, 128, 64, W2, nng, expw, scsh, fused + 0, fused + 64, FUSED_C);

  // Layer 3: 128 -> 128, no residual
  gemm_kernel<128, true, 0><<<GE, 256, 0, stream>>>(fused + 64, FUSED_C, 128, 64, W3, nng, expw, scsh, nullptr, part, 0);
  stats_reduce<<<128, 256, 0, stream>>>(part, GE, invME, g3, be3, scsh, 128);
  gemm_kernel<128, true, 1><<<GE, 256, 0, stream>>>(fused + 64, FUSED_C, 128, 64, W3, nng, expw, scsh, nullptr, fused + 128, FUSED_C);

  // Layer 4: 256 -> 128, residual (+feat3)
  gemm_kernel<128, true, 0><<<GE, 256, 0, stream>>>(fused + 128, FUSED_C, 256, 128, W4, nng, expw, scsh, nullptr, part, 0);
  stats_reduce<<<128, 256, 0, stream>>>(part, GE, invME, g4, be4, scsh, 128);
  gemm_kernel<128, true, 1><<<GE, 256, 0, stream>>>(fused + 128, FUSED_C, 256, 128, W4, nng, expw, scsh, fused + 128, fused + 256, FUSED_C);

  // Final conv: 384 -> 128 over points (direct GEMM)
  const int GF = BN / BLOCKM;                  // 128 blocks
  const float invMF = 1.f / (float)BN;
  gemm_kernel<128, false, 0><<<GF, 256, 0, stream>>>(fused, FUSED_C, FUSED_C, 0, Wo, nng, expw, scsh, nullptr, part, 0);
  stats_reduce<<<128, 256, 0, stream>>>(part, GF, invMF, go, beo, scsh, 128);
  gemm_kernel<128, false, 2><<<GF, 256, 0, stream>>>(fused, FUSED_C, FUSED_C, 0, Wo, nng, expw, scsh, nullptr, out, 128);
}